// ResidueGraphModel_56453050138694
// MI455X (gfx1250) — compile-verified
//
#include <hip/hip_runtime.h>
#include <hip/hip_bf16.h>

typedef __attribute__((ext_vector_type(2))) float v2f;
typedef __attribute__((ext_vector_type(8))) float v8f;

#define N_NODES 50000
#define N_EDGES 800000
#define HDIM 128
#define IN_DIM 512

// D = A(16x4 f32) * B(4x16 f32) + C   -> v_wmma_f32_16x16x4_f32
__device__ __forceinline__ v8f wmma4_f32(v2f a, v2f b, v8f c) {
  return __builtin_amdgcn_wmma_f32_16x16x4_f32(
      /*neg_a=*/false, a, /*neg_b=*/false, b,
      /*c_mod=*/(short)0, c, /*reuse_a=*/false, /*reuse_b=*/false);
}

// ---------------------------------------------------------------------------
// Kernel 1: X[50000,128] = feat[50000,512] @ W[512,128] + b
// grid = 3125 M-tiles, block = 256 (8 waves; wave w owns N-tile w)
// ---------------------------------------------------------------------------
__global__ void proj_gemm_kernel(const float* __restrict__ A,
                                 const float* __restrict__ W,
                                 const float* __restrict__ bias,
                                 float* __restrict__ X) {
  const int lane  = threadIdx.x & 31;
  const int wave  = threadIdx.x >> 5;
  const int mtile = blockIdx.x;
  const int mrow  = lane & 15;
  const int ncol  = (wave << 4) + (lane & 15);
  const int kgrp  = (lane >> 4) << 1;   // 0 or 2

  const float* arow = A + (size_t)(mtile * 16 + mrow) * IN_DIM + kgrp;
  const float* bcol = W + (size_t)kgrp * HDIM + ncol;

  v8f acc = {};
#pragma unroll 4
  for (int k0 = 0; k0 < IN_DIM; k0 += 4) {
    v2f a = *(const v2f*)(arow + k0);                 // A[m][k0+kgrp .. +1]
    v2f b;
    b.x = bcol[(size_t)k0 * HDIM];                    // W[k0+kgrp][n]
    b.y = bcol[(size_t)(k0 + 1) * HDIM];              // W[k0+kgrp+1][n]
    acc = wmma4_f32(a, b, acc);
  }
  const float bv = bias[ncol];
#pragma unroll
  for (int r = 0; r < 8; ++r) {
    const int row = ((lane >> 4) << 3) + r;           // C/D: M = r + 8*(lane>=16)
    X[(size_t)(mtile * 16 + row) * HDIM + ncol] = acc[r] + bv;
  }
}

// ---------------------------------------------------------------------------
// Zero the aggregation buffer
// ---------------------------------------------------------------------------
__global__ void zero_kernel(float* __restrict__ p, int n) {
  const int i = blockIdx.x * blockDim.x + threadIdx.x;
  if (i < n) p[i] = 0.0f;
}

// ---------------------------------------------------------------------------
// Kernel 2: per edge e: msg = relu(X[src[e]] + emb[attr[e]]); agg[dst[e]] += msg
// 1 wave per edge, lane handles a float4 chunk of H=128.
// ---------------------------------------------------------------------------
__global__ void edge_scatter_kernel(const float* __restrict__ X,
                                    const float* __restrict__ emb,
                                    const int* __restrict__ src,
                                    const int* __restrict__ dst,
                                    const int* __restrict__ eattr,
                                    float* __restrict__ agg) {
  const int lane = threadIdx.x & 31;
  const int e    = blockIdx.x * 8 + (threadIdx.x >> 5);
  if (e >= N_EDGES) return;
  const int s = src[e];
  const int d = dst[e];
  const int a = eattr[e];

  const float4 xv = *(const float4*)(X + (size_t)s * HDIM + lane * 4);
  const float4 ev = *(const float4*)(emb + (size_t)a * HDIM + lane * 4);
  const float m0 = fmaxf(xv.x + ev.x, 0.0f);
  const float m1 = fmaxf(xv.y + ev.y, 0.0f);
  const float m2 = fmaxf(xv.z + ev.z, 0.0f);
  const float m3 = fmaxf(xv.w + ev.w, 0.0f);

  float* ap = agg + (size_t)d * HDIM + lane * 4;
  atomicAdd(ap + 0, m0);   // global_atomic_add_f32, L2-resident target
  atomicAdd(ap + 1, m1);
  atomicAdd(ap + 2, m2);
  atomicAdd(ap + 3, m3);
}

// ---------------------------------------------------------------------------
// Kernel 3: h = X + agg;  X = relu(h@W1 + b1)@W2 + b2 + X   (in place)
// grid = 3125 M-tiles, block = 256 (8 waves = 8 N-tiles). LDS pitch 132 floats.
// ---------------------------------------------------------------------------
__global__ void mlp_kernel(const float* __restrict__ agg,
                           const float* __restrict__ W1,
                           const float* __restrict__ b1,
                           const float* __restrict__ W2,
                           const float* __restrict__ b2,
                           float* __restrict__ X) {
  __shared__ float sh_h[16 * 132];
  __shared__ float sh_t[16 * 132];

  const int lane  = threadIdx.x & 31;
  const int wave  = threadIdx.x >> 5;
  const int mtile = blockIdx.x;
  const size_t base = (size_t)mtile * 16 * HDIM;

  // phase 0: stage h = x + agg into LDS (each thread: 8 contiguous floats)
  {
    const int row = threadIdx.x >> 4;          // (t*8)/128
    const int col = (threadIdx.x & 15) * 8;
    const float4* xp = (const float4*)(X + base + (size_t)row * HDIM + col);
    const float4* gp = (const float4*)(agg + base + (size_t)row * HDIM + col);
    const float4 x0 = xp[0], x1 = xp[1];
    const float4 g0 = gp[0], g1 = gp[1];
    float* o = sh_h + row * 132 + col;
    o[0] = x0.x + g0.x; o[1] = x0.y + g0.y; o[2] = x0.z + g0.z; o[3] = x0.w + g0.w;
    o[4] = x1.x + g1.x; o[5] = x1.y + g1.y; o[6] = x1.z + g1.z; o[7] = x1.w + g1.w;
  }
  __syncthreads();

  const int mrow = lane & 15;
  const int ncol = (wave << 4) + (lane & 15);
  const int kgrp = (lane >> 4) << 1;

  // phase 1: t = relu(h @ W1 + b1) -> LDS
  {
    v8f acc = {};
    const float* ar = sh_h + mrow * 132 + kgrp;
    const float* bc = W1 + (size_t)kgrp * HDIM + ncol;
#pragma unroll 4
    for (int k0 = 0; k0 < HDIM; k0 += 4) {
      v2f a = *(const v2f*)(ar + k0);
      v2f b;
      b.x = bc[(size_t)k0 * HDIM];
      b.y = bc[(size_t)(k0 + 1) * HDIM];
      acc = wmma4_f32(a, b, acc);
    }
    const float bv = b1[ncol];
#pragma unroll
    for (int r = 0; r < 8; ++r) {
      const int row = ((lane >> 4) << 3) + r;
      sh_t[row * 132 + ncol] = fmaxf(acc[r] + bv, 0.0f);
    }
  }
  __syncthreads();

  // phase 2: X = t @ W2 + b2 + X
  {
    v8f acc = {};
    const float* ar = sh_t + mrow * 132 + kgrp;
    const float* bc = W2 + (size_t)kgrp * HDIM + ncol;
#pragma unroll 4
    for (int k0 = 0; k0 < HDIM; k0 += 4) {
      v2f a = *(const v2f*)(ar + k0);
      v2f b;
      b.x = bc[(size_t)k0 * HDIM];
      b.y = bc[(size_t)(k0 + 1) * HDIM];
      acc = wmma4_f32(a, b, acc);
    }
    const float bv = b2[ncol];
#pragma unroll
    for (int r = 0; r < 8; ++r) {
      const int row = ((lane >> 4) << 3) + r;
      const size_t idx = base + (size_t)row * HDIM + ncol;
      X[idx] = acc[r] + bv + X[idx];
    }
  }
}

// ---------------------------------------------------------------------------
// Kernel 4: LayerNorm over H=128, 1 wave per row (lane = float4 chunk)
// ---------------------------------------------------------------------------
__global__ void layernorm_kernel(float* __restrict__ X,
                                 const float* __restrict__ gamma,
                                 const float* __restrict__ beta) {
  const int lane = threadIdx.x & 31;
  const int row  = blockIdx.x * 8 + (threadIdx.x >> 5);
  if (row >= N_NODES) return;

  const float4 v = *(const float4*)(X + (size_t)row * HDIM + lane * 4);
  float s = v.x + v.y + v.z + v.w;
  float q = v.x * v.x + v.y * v.y + v.z * v.z + v.w * v.w;
#pragma unroll
  for (int off = 16; off >= 1; off >>= 1) {
    s += __shfl_xor(s, off, 32);
    q += __shfl_xor(q, off, 32);
  }
  const float mean = s * (1.0f / 128.0f);
  const float var  = q * (1.0f / 128.0f) - mean * mean;
  const float inv  = rsqrtf(var + 1e-5f);

  const float4 g  = *(const float4*)(gamma + lane * 4);
  const float4 bt = *(const float4*)(beta + lane * 4);
  float4 o;
  o.x = g.x * (v.x - mean) * inv + bt.x;
  o.y = g.y * (v.y - mean) * inv + bt.y;
  o.z = g.z * (v.z - mean) * inv + bt.z;
  o.w = g.w * (v.w - mean) * inv + bt.w;
  *(float4*)(X + (size_t)row * HDIM + lane * 4) = o;
}

// ---------------------------------------------------------------------------
extern "C" void kernel_launch(void* const* d_in, const int* in_sizes, int n_in,
                              void* d_out, int out_size, void* d_ws, size_t ws_size,
                              hipStream_t stream) {
  (void)in_sizes; (void)n_in; (void)out_size; (void)ws_size;
  const float* feat  = (const float*)d_in[0];
  const float* projW = (const float*)d_in[1];
  const float* projb = (const float*)d_in[2];
  const float* eemb  = (const float*)d_in[3];
  const float* W1    = (const float*)d_in[4];
  const float* b1    = (const float*)d_in[5];
  const float* W2    = (const float*)d_in[6];
  const float* b2    = (const float*)d_in[7];
  const float* gam   = (const float*)d_in[8];
  const float* bet   = (const float*)d_in[9];
  const int*   eidx  = (const int*)d_in[10];   // [2, N_EDGES] int32
  const int*   eatt  = (const int*)d_in[11];   // [N_EDGES, 1] int32

  float* X   = (float*)d_out;   // node features live in d_out (25.6 MB)
  float* agg = (float*)d_ws;    // scatter-sum accumulator (25.6 MB)

  proj_gemm_kernel<<<N_NODES / 16, 256, 0, stream>>>(feat, projW, projb, X);

  for (int l = 0; l < 3; ++l) {
    zero_kernel<<<(N_NODES * HDIM + 255) / 256, 256, 0, stream>>>(agg, N_NODES * HDIM);
    edge_scatter_kernel<<<(N_EDGES + 7) / 8, 256, 0, stream>>>(
        X, eemb, eidx, eidx + N_EDGES, eatt, agg);
    mlp_kernel<<<N_NODES / 16, 256, 0, stream>>>(
        agg, W1 + (size_t)l * HDIM * HDIM, b1 + (size_t)l * HDIM,
        W2 + (size_t)l * HDIM * HDIM, b2 + (size_t)l * HDIM, X);
  }

  layernorm_kernel<<<N_NODES / 8, 256, 0, stream>>>(X, gam, bet);
}